// GNNModel_41274635715016
// MI455X (gfx1250) — compile-verified
//
#include <hip/hip_runtime.h>
#include <hip/hip_bf16.h>

typedef __attribute__((ext_vector_type(2))) float v2f;
typedef __attribute__((ext_vector_type(8))) float v8f;
typedef __attribute__((ext_vector_type(4))) unsigned int u32x4;
typedef __attribute__((ext_vector_type(8))) int i32x8;
typedef __attribute__((ext_vector_type(4))) int i32x4;

#define DIMD   256
#define HID    256
#define NCLS   40
#define FBINS  64     // F(4) * NUM_BINS(16)
#define SCAN_T 256
#define TILE_M 32

static __device__ __forceinline__ v8f wmma_f32(v2f a, v2f b, v8f c) {
    return __builtin_amdgcn_wmma_f32_16x16x4_f32(false, a, false, b, (short)0, c,
                                                 false, false);
}

// ---------------- setup / scatter kernels ----------------

__global__ void init_ws_kernel(int* __restrict__ pos, int* __restrict__ aggI,
                               int N, int E) {
    int i = blockIdx.x * 256 + threadIdx.x;
    if (i < N) pos[i] = E;
    if (i < N * FBINS) aggI[i] = 0;
}

__global__ void pos_kernel(const int* __restrict__ ei, int* __restrict__ pos, int E) {
    int e = blockIdx.x * 256 + threadIdx.x;
    if (e < E) atomicMin(&pos[ei[2 * e]], e);
}

static __device__ __forceinline__ int block_incl_scan(int v, int t, int* sdata) {
    sdata[t] = v;
    __syncthreads();
    for (int off = 1; off < SCAN_T; off <<= 1) {
        int x = (t >= off) ? sdata[t - off] : 0;
        __syncthreads();
        sdata[t] += x;
        __syncthreads();
    }
    return sdata[t];
}

// per-block sums of first-occurrence flags over 1024 edges/block
__global__ void edge_blocksums_kernel(const int* __restrict__ ei,
                                      const int* __restrict__ pos,
                                      int* __restrict__ blk, int E) {
    __shared__ int sdata[SCAN_T];
    int b = blockIdx.x, t = threadIdx.x;
    int base = b * 1024 + t * 4;
    int s = 0;
    for (int j = 0; j < 4; ++j) {
        int e = base + j;
        if (e < E) s += (pos[ei[2 * e]] == e) ? 1 : 0;
    }
    sdata[t] = s;
    __syncthreads();
    for (int off = 128; off > 0; off >>= 1) {
        if (t < off) sdata[t] += sdata[t + off];
        __syncthreads();
    }
    if (t == 0) blk[b] = sdata[0];
}

// single-block exclusive scan of `data[0..n)`; writes grand total to *totalOut
__global__ void scan_exclusive_kernel(int* __restrict__ data, int n,
                                      int* __restrict__ totalOut) {
    __shared__ int sdata[SCAN_T];
    __shared__ int running;
    int t = threadIdx.x;
    if (t == 0) running = 0;
    __syncthreads();
    for (int base = 0; base < n; base += SCAN_T) {
        int v = (base + t < n) ? data[base + t] : 0;
        int incl = block_incl_scan(v, t, sdata);
        int r = running;
        if (base + t < n) data[base + t] = r + (incl - v);
        __syncthreads();
        if (t == SCAN_T - 1) running = r + sdata[SCAN_T - 1];
        __syncthreads();
    }
    if (t == 0 && totalOut) *totalOut = running;
}

// write rank for appearing nodes: rank[node] = global exclusive prefix of
// first-occurrence flags at the node's first-occurrence edge
__global__ void edge_rank_kernel(const int* __restrict__ ei,
                                 const int* __restrict__ pos,
                                 const int* __restrict__ blk,
                                 int* __restrict__ rank, int E) {
    __shared__ int sdata[SCAN_T];
    int b = blockIdx.x, t = threadIdx.x;
    int base = b * 1024 + t * 4;
    int f[4], nd[4], s = 0;
    for (int j = 0; j < 4; ++j) {
        int e = base + j;
        if (e < E) {
            nd[j] = ei[2 * e];
            f[j] = (pos[nd[j]] == e) ? 1 : 0;
        } else {
            nd[j] = 0;
            f[j] = 0;
        }
        s += f[j];
    }
    int incl = block_incl_scan(s, t, sdata);
    int off = blk[b] + (incl - s);
    for (int j = 0; j < 4; ++j) {
        if (f[j]) { rank[nd[j]] = off; off++; }
    }
}

__global__ void node_blocksums_kernel(const int* __restrict__ pos,
                                      int* __restrict__ blk, int N, int E) {
    __shared__ int sdata[SCAN_T];
    int b = blockIdx.x, t = threadIdx.x;
    int base = b * 1024 + t * 4;
    int s = 0;
    for (int j = 0; j < 4; ++j) {
        int n = base + j;
        if (n < N) s += (pos[n] == E) ? 1 : 0;
    }
    sdata[t] = s;
    __syncthreads();
    for (int off = 128; off > 0; off >>= 1) {
        if (t < off) sdata[t] += sdata[t + off];
        __syncthreads();
    }
    if (t == 0) blk[b] = sdata[0];
}

__global__ void node_rank_kernel(const int* __restrict__ pos,
                                 const int* __restrict__ blk,
                                 const int* __restrict__ Kapp,
                                 int* __restrict__ rank, int N, int E) {
    __shared__ int sdata[SCAN_T];
    int b = blockIdx.x, t = threadIdx.x;
    int base = b * 1024 + t * 4;
    int f[4], s = 0;
    for (int j = 0; j < 4; ++j) {
        int n = base + j;
        f[j] = (n < N && pos[n] == E) ? 1 : 0;
        s += f[j];
    }
    int incl = block_incl_scan(s, t, sdata);
    int off = Kapp[0] + blk[b] + (incl - s);
    for (int j = 0; j < 4; ++j) {
        int n = base + j;
        if (f[j]) { rank[n] = off; off++; }
    }
}

// histogram: aggI[rank[seg[seg[e]]]][f*16+bin]++
__global__ void scatter_kernel(const int* __restrict__ ei,
                               const int* __restrict__ ef,
                               const int* __restrict__ rank,
                               int* __restrict__ aggI, int E) {
    int e = blockIdx.x * 256 + threadIdx.x;
    if (e >= E) return;
    int a = ei[2 * e];          // seg[e], a node id (< N <= E)
    int nd = ei[2 * a];         // seg[seg[e]]
    int m = rank[nd];
    int base = m * FBINS;
#pragma unroll
    for (int f = 0; f < 4; ++f) {
        int bin = ef[e * 4 + f];
        atomicAdd(&aggI[base + f * 16 + bin], 1);
    }
}

// ---------------- WMMA kernels ----------------

// Fused: 32-row tile per block. TDM stages the x tile into LDS; 8 waves
// compute h = relu(x@W1+b1) with 2 M-tiles each (B-frag reuse); h overwrites
// the same LDS buffer; 6 waves then compute out = h @ W2[0:256] + b2.
__global__ __launch_bounds__(256) void gemm_fused_kernel(
    const float* __restrict__ x, const float* __restrict__ W1,
    const float* __restrict__ b1, const float* __restrict__ W2,
    const float* __restrict__ b2, float* __restrict__ out, int N) {
    __shared__ float sbuf[TILE_M * DIMD];   // 32 KB: x tile, then h tile
    int rowBase = blockIdx.x * TILE_M;
    int t = threadIdx.x;

    // --- stage 32x256 fp32 tile via Tensor Data Mover (wave 0 issues) ---
    if (t < 32) {
        unsigned lds_off = (unsigned)(uintptr_t)(&sbuf[0]);
        unsigned long long ga =
            (unsigned long long)(uintptr_t)(x + (size_t)rowBase * DIMD);
        u32x4 g0;
        g0.x = 1u;                                   // count=1, user mode
        g0.y = lds_off;                              // lds_addr
        g0.z = (unsigned)(ga & 0xFFFFFFFFu);         // global_addr[31:0]
        g0.w = (unsigned)((ga >> 32) & 0x1FFFFFFu)   // global_addr[56:32]
               | (2u << 30);                         // type=2 (image)
        unsigned remRows = (unsigned)(N - rowBase);
        i32x8 g1;
        g1[0] = (int)(2u << 16);                             // data_size=4B
        g1[1] = (int)((unsigned)DIMD << 16);                 // tensor_dim0[15:0]
        g1[2] = (int)((remRows & 0xFFFFu) << 16);            // dim0 hi | dim1 lo
        g1[3] = (int)((remRows >> 16) | ((unsigned)DIMD << 16)); // dim1 hi | tile_dim0
        g1[4] = TILE_M;                                      // tile_dim1 (rows)
        g1[5] = DIMD;                                        // dim0_stride lo32
        g1[6] = 0;
        g1[7] = 0;
        i32x4 gz4 = {0, 0, 0, 0};
        i32x8 gz8 = {0, 0, 0, 0, 0, 0, 0, 0};
        __builtin_amdgcn_tensor_load_to_lds(g0, g1, gz4, gz4, gz8, 0);
        __builtin_amdgcn_s_wait_tensorcnt(0);
    }
    __syncthreads();

    int lane = t & 31, wave = t >> 5;
    int m = lane & 15;                  // A row index within tile
    int kb = (lane >> 4) << 1;          // K sub-offset: 0 or 2
    int n = lane & 15;                  // B/C column within tile
    int mh = (lane >> 4) * 8;           // C row half offset

    // phase 1: wave w -> h cols [32w, 32w+32), rows 0..31 (two 16-row tiles)
    int nb0 = wave * 32, nb1 = nb0 + 16;
    v8f c00 = {}, c01 = {}, c10 = {}, c11 = {};
    for (int k0 = 0; k0 < DIMD; k0 += 4) {
        v2f a0, a1, b0, b1v;
        a0.x = sbuf[m * DIMD + k0 + kb];
        a0.y = sbuf[m * DIMD + k0 + kb + 1];
        a1.x = sbuf[(16 + m) * DIMD + k0 + kb];
        a1.y = sbuf[(16 + m) * DIMD + k0 + kb + 1];
        b0.x = W1[(k0 + kb) * HID + nb0 + n];
        b0.y = W1[(k0 + kb + 1) * HID + nb0 + n];
        b1v.x = W1[(k0 + kb) * HID + nb1 + n];
        b1v.y = W1[(k0 + kb + 1) * HID + nb1 + n];
        c00 = wmma_f32(a0, b0, c00);
        c01 = wmma_f32(a0, b1v, c01);
        c10 = wmma_f32(a1, b0, c10);
        c11 = wmma_f32(a1, b1v, c11);
    }
    __syncthreads();   // everyone done reading x tile; reuse sbuf for h
    {
        float bias0 = b1[nb0 + n], bias1 = b1[nb1 + n];
#pragma unroll
        for (int i = 0; i < 8; ++i) {
            int r0 = i + mh, r1 = 16 + i + mh;
            float v00 = c00[i] + bias0, v01 = c01[i] + bias1;
            float v10 = c10[i] + bias0, v11 = c11[i] + bias1;
            sbuf[r0 * HID + nb0 + n] = v00 > 0.0f ? v00 : 0.0f;
            sbuf[r0 * HID + nb1 + n] = v01 > 0.0f ? v01 : 0.0f;
            sbuf[r1 * HID + nb0 + n] = v10 > 0.0f ? v10 : 0.0f;
            sbuf[r1 * HID + nb1 + n] = v11 > 0.0f ? v11 : 0.0f;
        }
    }
    __syncthreads();

    // phase 2: 6 waves -> (2 row-tiles) x (3 n-tiles of 16, pad 48 keep 40)
    if (wave < 6) {
        int rt = wave / 3, nt = wave - rt * 3;
        int rowOff = rt * 16;
        int col = nt * 16 + n;
        bool valid = (col < NCLS);
        v8f cc = {};
        for (int k0 = 0; k0 < HID; k0 += 4) {
            v2f a, b;
            a.x = sbuf[(rowOff + m) * HID + k0 + kb];
            a.y = sbuf[(rowOff + m) * HID + k0 + kb + 1];
            b.x = valid ? W2[(k0 + kb) * NCLS + col] : 0.0f;
            b.y = valid ? W2[(k0 + kb + 1) * NCLS + col] : 0.0f;
            cc = wmma_f32(a, b, cc);
        }
        if (valid) {
            float bb = b2[col];
#pragma unroll
            for (int i = 0; i < 8; ++i) {
                int row = rowBase + rowOff + i + mh;
                out[row * NCLS + col] = cc[i] + bb;
            }
        }
    }
}

// out += agg(float counts) @ W2[256:320]. One wave per 16-row tile, 3 n-tiles.
__global__ __launch_bounds__(32) void gemm_agg_kernel(
    const int* __restrict__ aggI, const float* __restrict__ W2,
    float* __restrict__ out) {
    int rowBase = blockIdx.x * 16;
    int lane = threadIdx.x & 31;
    int m = lane & 15;
    int kb = (lane >> 4) << 1;
    int n = lane & 15;
    int mh = (lane >> 4) * 8;
    const float* W2b = W2 + HID * NCLS;   // rows 256..319
    int c2col = 32 + n;
    bool v2ok = (c2col < NCLS);

    v8f c0 = {}, c1 = {}, c2 = {};
    for (int k0 = 0; k0 < FBINS; k0 += 4) {
        int r0 = k0 + kb, r1 = k0 + kb + 1;
        v2f a;
        a.x = (float)aggI[(rowBase + m) * FBINS + r0];
        a.y = (float)aggI[(rowBase + m) * FBINS + r1];
        v2f b0, b1v, b2v;
        b0.x = W2b[r0 * NCLS + n];
        b0.y = W2b[r1 * NCLS + n];
        b1v.x = W2b[r0 * NCLS + 16 + n];
        b1v.y = W2b[r1 * NCLS + 16 + n];
        b2v.x = v2ok ? W2b[r0 * NCLS + c2col] : 0.0f;
        b2v.y = v2ok ? W2b[r1 * NCLS + c2col] : 0.0f;
        c0 = wmma_f32(a, b0, c0);
        c1 = wmma_f32(a, b1v, c1);
        c2 = wmma_f32(a, b2v, c2);
    }
#pragma unroll
    for (int i = 0; i < 8; ++i) {
        int row = rowBase + i + mh;
        out[row * NCLS + n] += c0[i];
        out[row * NCLS + 16 + n] += c1[i];
        if (v2ok) out[row * NCLS + c2col] += c2[i];
    }
}

// ---------------- host launcher ----------------

extern "C" void kernel_launch(void* const* d_in, const int* in_sizes, int n_in,
                              void* d_out, int out_size, void* d_ws, size_t ws_size,
                              hipStream_t stream) {
    const float* x  = (const float*)d_in[0];
    const int*   ei = (const int*)d_in[1];
    const int*   ef = (const int*)d_in[2];
    const float* W1 = (const float*)d_in[3];
    const float* b1 = (const float*)d_in[4];
    const float* W2 = (const float*)d_in[5];
    const float* b2 = (const float*)d_in[6];
    float* out = (float*)d_out;

    const int N = in_sizes[0] / DIMD;   // 100000
    const int E = in_sizes[1] / 2;      // 3200000

    // workspace layout (ints)
    int* pos  = (int*)d_ws;                  // N
    int* rank = pos + N;                     // N
    int* aggI = rank + N;                    // N*64
    int* blkE = aggI + (size_t)N * FBINS;    // NB_E
    const int NB_E = (E + 1023) / 1024;
    int* blkN = blkE + NB_E;                 // NB_N
    const int NB_N = (N + 1023) / 1024;
    int* Kapp = blkN + NB_N;                 // 1

    // 1. init pos=E, aggI=0
    init_ws_kernel<<<((size_t)N * FBINS + 255) / 256, 256, 0, stream>>>(pos, aggI, N, E);
    // 2. pos[n] = min first-occurrence edge index
    pos_kernel<<<(E + 255) / 256, 256, 0, stream>>>(ei, pos, E);
    // 3-5. rank of appearing nodes via prefix sum of first-occurrence flags
    edge_blocksums_kernel<<<NB_E, SCAN_T, 0, stream>>>(ei, pos, blkE, E);
    scan_exclusive_kernel<<<1, SCAN_T, 0, stream>>>(blkE, NB_E, Kapp);
    edge_rank_kernel<<<NB_E, SCAN_T, 0, stream>>>(ei, pos, blkE, rank, E);
    // 6-8. rank of non-appearing nodes (pos==E), ordered by node index
    node_blocksums_kernel<<<NB_N, SCAN_T, 0, stream>>>(pos, blkN, N, E);
    scan_exclusive_kernel<<<1, SCAN_T, 0, stream>>>(blkN, NB_N, nullptr);
    node_rank_kernel<<<NB_N, SCAN_T, 0, stream>>>(pos, blkN, Kapp, rank, N, E);
    // 9. one-hot histogram scatter
    scatter_kernel<<<(E + 255) / 256, 256, 0, stream>>>(ei, ef, rank, aggI, E);
    // 10. fused h = relu(xW1+b1); out = h @ W2[:256] + b2   (WMMA f32 + TDM)
    gemm_fused_kernel<<<N / TILE_M, 256, 0, stream>>>(x, W1, b1, W2, b2, out, N);
    // 11. out += agg @ W2[256:]                             (WMMA f32)
    gemm_agg_kernel<<<N / 16, 32, 0, stream>>>(aggI, W2, out);
}